// ParticlesNetwork_55825984913902
// MI455X (gfx1250) — compile-verified
//
#include <hip/hip_runtime.h>
#include <math.h>

#define RBINS 3
#define TBINS 16
#define GBINS 48
#define NPART 512
#define BATCH 2
#define MROWS (BATCH*NPART)   /* 1024 */
#define RADIUS_F 40.0f
#define SHIFT_F 0.2f
#define CORR_F (1.0f/128.0f)
#define PI_F 3.14159265358979323846f
#define TWO_PI_F 6.28318530717958647692f

typedef __attribute__((ext_vector_type(2))) float v2f;
typedef __attribute__((ext_vector_type(8))) float v8f;

// ---------------------------------------------------------------------------
// 1) lift fluid_feats (v1, p1, v0_enc, p0_enc) through both rho1->reg bases
// ---------------------------------------------------------------------------
__global__ void k_init_feats(const float* __restrict__ p0_enc,
                             const float* __restrict__ v0_enc,
                             const float* __restrict__ p0,
                             const float* __restrict__ v0,
                             const float* __restrict__ acc,
                             const float* __restrict__ lift_c0,
                             const float* __restrict__ lift_d0,
                             float* __restrict__ p1,
                             float* __restrict__ Xc,
                             float* __restrict__ Xd)
{
    int idx = blockIdx.x * blockDim.x + threadIdx.x;
    if (idx >= MROWS * 16) return;
    int c  = idx & 15;
    int bi = idx >> 4;
    float fx, fy;
    if (c == 0) {                       // v1 = v0 + a
        fx = v0[bi*2+0] + acc[bi*2+0];
        fy = v0[bi*2+1] + acc[bi*2+1];
    } else if (c == 1) {                // p1 = p0 + v0 + a/2
        fx = p0[bi*2+0] + v0[bi*2+0] + 0.5f*acc[bi*2+0];
        fy = p0[bi*2+1] + v0[bi*2+1] + 0.5f*acc[bi*2+1];
        p1[bi*2+0] = fx; p1[bi*2+1] = fy;
    } else if (c <= 8) {                // v0_enc channels
        int t = c - 2;
        fx = v0_enc[(bi*7+t)*2+0]; fy = v0_enc[(bi*7+t)*2+1];
    } else {                            // p0_enc channels
        int t = c - 9;
        fx = p0_enc[(bi*7+t)*2+0]; fy = p0_enc[(bi*7+t)*2+1];
    }
    float ca = lift_c0[0], cb = lift_c0[1];
    float da = lift_d0[0], db = lift_d0[1];
    #pragma unroll
    for (int m = 0; m < 8; ++m) {
        float th = (float)m * (PI_F/4.0f);
        float ct = cosf(th), st = sinf(th);
        // basis[m] = (a*cos - b*sin, a*sin + b*cos)
        Xc[bi*128 + c*8 + m] = fx*(ca*ct - cb*st) + fy*(ca*st + cb*ct);
        Xd[bi*128 + c*8 + m] = fx*(da*ct - db*st) + fy*(da*st + db*ct);
    }
}

// ---------------------------------------------------------------------------
// 2) pair-bin weights: denominator then packed (4 bins, 4 weights) per pair
// ---------------------------------------------------------------------------
__global__ void k_denom(const float* __restrict__ p1,
                        const float* __restrict__ mask,
                        float* __restrict__ denom)
{
    int bi = blockIdx.x * blockDim.x + threadIdx.x;
    if (bi >= MROWS) return;
    int b = bi / NPART;
    float px = p1[bi*2+0], py = p1[bi*2+1];
    float s = 0.f;
    for (int j = 0; j < NPART; ++j) {
        float dx = p1[(b*NPART+j)*2+0] - px;
        float dy = p1[(b*NPART+j)*2+1] - py;
        float d2 = dx*dx + dy*dy + 1e-12f;
        float d  = sqrtf(d2) * (1.0f/RADIUS_F);
        float w  = fmaxf(1.0f - d*d, 0.0f);
        s += w*w*w * mask[b*NPART+j];
    }
    denom[bi] = s + 1e-6f;
}

__global__ void k_pairs(const float* __restrict__ p1,
                        const float* __restrict__ mask,
                        const float* __restrict__ denom,
                        unsigned int* __restrict__ gpk,
                        float4* __restrict__ w4)
{
    int idx = blockIdx.x * blockDim.x + threadIdx.x;
    if (idx >= MROWS * NPART) return;
    int j  = idx % NPART;
    int bi = idx / NPART;
    int b  = bi / NPART;
    int i  = bi % NPART;
    float dx = p1[(b*NPART+j)*2+0] - p1[bi*2+0];
    float dy = p1[(b*NPART+j)*2+1] - p1[bi*2+1];
    float d2 = dx*dx + dy*dy + 1e-12f;
    float d  = sqrtf(d2) * (1.0f/RADIUS_F);
    float win = fmaxf(1.0f - d*d, 0.0f);
    win = win*win*win;
    float wb = win * mask[b*NPART+j] / denom[bi];
    float rc = fminf(fmaxf(d*(float)RBINS - 0.5f, 0.0f), (float)(RBINS-1));
    float r0 = floorf(rc);
    float wr = rc - r0;
    int r0i = (int)r0;
    int r1i = r0i + 1 > RBINS-1 ? RBINS-1 : r0i + 1;
    float rx = (i == j) ? 1.0f : dx;
    float ry = (i == j) ? 0.0f : dy;
    float th = atan2f(ry, rx);
    if (th < 0.f) th += TWO_PI_F;
    float tc = th * ((float)TBINS / TWO_PI_F);
    float t0 = floorf(tc);
    float wt = tc - t0;
    int t0i = ((int)t0) & 15;
    int t1i = (t0i + 1) & 15;
    int g0 = r0i*TBINS + t0i, g1 = r0i*TBINS + t1i;
    int g2 = r1i*TBINS + t0i, g3 = r1i*TBINS + t1i;
    gpk[idx] = (unsigned)g0 | ((unsigned)g1 << 8) |
               ((unsigned)g2 << 16) | ((unsigned)g3 << 24);
    w4[idx] = make_float4((1.f-wr)*(1.f-wt)*wb, (1.f-wr)*wt*wb,
                          wr*(1.f-wt)*wb,       wr*wt*wb);
}

// ---------------------------------------------------------------------------
// 3) gather: F[bi][(g,c,n)] = sum_j w(i,j,g) * X[b,j,(c,n)]  (LDS accumulate)
// ---------------------------------------------------------------------------
__global__ void k_gather(const float* __restrict__ X,
                         const unsigned int* __restrict__ gpk,
                         const float4* __restrict__ w4,
                         float* __restrict__ F, int CN)
{
    __shared__ float Flds[GBINS * 128];
    __shared__ unsigned int gs[128];
    __shared__ float4 wsh[128];
    int bi  = blockIdx.x;
    int b   = bi / NPART;
    int tid = threadIdx.x;
    for (int idx = tid; idx < GBINS*CN; idx += 128) Flds[idx] = 0.f;
    for (int j0 = 0; j0 < NPART; j0 += 128) {
        __syncthreads();
        gs[tid]  = gpk[bi*NPART + j0 + tid];
        wsh[tid] = w4 [bi*NPART + j0 + tid];
        __syncthreads();
        if (tid < CN) {
            for (int jj = 0; jj < 128; ++jj) {
                float4 w = wsh[jj];
                if (w.x == 0.f && w.y == 0.f && w.z == 0.f && w.w == 0.f) continue;
                float x = X[(size_t)(b*NPART + j0 + jj)*CN + tid];
                unsigned int g = gs[jj];
                Flds[((g      ) & 255u)*CN + tid] += w.x * x;
                Flds[((g >>  8) & 255u)*CN + tid] += w.y * x;
                Flds[((g >> 16) & 255u)*CN + tid] += w.z * x;
                Flds[((g >> 24) & 255u)*CN + tid] += w.w * x;
            }
        }
    }
    __syncthreads();
    for (int idx = tid; idx < GBINS*CN; idx += 128)
        F[(size_t)bi*GBINS*CN + idx] = Flds[idx];
}

// ---------------------------------------------------------------------------
// 4) weight rearrangement: Kmat[(g,c,n)][(o,m)] and Dmat[(c,n)][(o,m)]
// ---------------------------------------------------------------------------
__global__ void k_build_kmat(const float* __restrict__ Wc,
                             float* __restrict__ Kmat, int Cin, int Cout)
{
    int ncol = Cout*8;
    int total = GBINS*Cin*8*ncol;
    int idx = blockIdx.x * blockDim.x + threadIdx.x;
    if (idx >= total) return;
    int col = idx % ncol;
    int k   = idx / ncol;
    int o = col >> 3, m = col & 7;
    int n = k & 7, c = (k >> 3) % Cin, g = k / (8*Cin);
    int r = g >> 4, t = g & 15;
    int kk = (n - m) & 7;
    int tt = (t - 2*m) & 15;
    Kmat[idx] = Wc[(((o*Cin + c)*8 + kk)*RBINS + r)*TBINS + tt];
}

__global__ void k_build_kmat4(const float* __restrict__ Wc,
                              const float* __restrict__ proj,
                              float* __restrict__ Kmat, int Cin, int Cout)
{
    int total = GBINS*Cin*8*16;
    int idx = blockIdx.x * blockDim.x + threadIdx.x;
    if (idx >= total) return;
    int col = idx & 15;
    int k   = idx >> 4;
    float val = 0.f;
    if (col < Cout*2) {
        int o = col >> 1, aa = col & 1;
        int n = k & 7, c = (k >> 3) % Cin, g = k / (8*Cin);
        int r = g >> 4, t = g & 15;
        float pa = proj[0], pb = proj[1];
        #pragma unroll
        for (int m = 0; m < 8; ++m) {
            int kk = (n - m) & 7;
            int tt = (t - 2*m) & 15;
            float w = Wc[(((o*Cin + c)*8 + kk)*RBINS + r)*TBINS + tt];
            float th = (float)m * (PI_F/4.0f);
            float ct = cosf(th), st = sinf(th);
            float bv = aa ? (pa*st + pb*ct) : (pa*ct - pb*st);
            val += w * bv;
        }
    }
    Kmat[idx] = val;
}

__global__ void k_build_dmat(const float* __restrict__ Wd,
                             float* __restrict__ Dmat, int Cin, int Cout)
{
    int ncol = Cout*8;
    int total = Cin*8*ncol;
    int idx = blockIdx.x * blockDim.x + threadIdx.x;
    if (idx >= total) return;
    int col = idx % ncol;
    int k   = idx / ncol;
    int o = col >> 3, m = col & 7;
    int n = k & 7, c = k >> 3;
    Dmat[idx] = Wd[(o*Cin + c)*8 + ((m - n) & 7)];
}

__global__ void k_build_dmat4(const float* __restrict__ Wd,
                              const float* __restrict__ proj,
                              float* __restrict__ Dmat, int Cin, int Cout)
{
    int total = Cin*8*16;
    int idx = blockIdx.x * blockDim.x + threadIdx.x;
    if (idx >= total) return;
    int col = idx & 15;
    int k   = idx >> 4;
    float val = 0.f;
    if (col < Cout*2) {
        int o = col >> 1, aa = col & 1;
        int n = k & 7, c = k >> 3;
        float pa = proj[0], pb = proj[1];
        #pragma unroll
        for (int m = 0; m < 8; ++m) {
            float w = Wd[(o*Cin + c)*8 + ((m - n) & 7)];
            float th = (float)m * (PI_F/4.0f);
            float ct = cosf(th), st = sinf(th);
            float bv = aa ? (pa*st + pb*ct) : (pa*ct - pb*st);
            val += w * bv;
        }
    }
    Dmat[idx] = val;
}

// ---------------------------------------------------------------------------
// 5) WMMA GEMM: out = A1@W1 (+ A2@W2) (+ resid), f32 16x16x4 WMMA.
//    One wave computes a 16 x (16*NT) output strip: the A-fragment (the large
//    F matrix) is loaded once per k-step and feeds NT WMMAs, cutting A traffic
//    by NT and raising matrix-op density per issue slot.
// ---------------------------------------------------------------------------
template<int NT>
__global__ void k_gemm(const float* __restrict__ A1, const float* __restrict__ W1, int K1,
                       const float* __restrict__ A2, const float* __restrict__ W2, int K2,
                       const float* __restrict__ resid, float* __restrict__ out,
                       int ldW, int ncolsValid, int colOff, int ldOut)
{
    int lane = threadIdx.x;
    int row0 = blockIdx.x * 16;
    int col0 = blockIdx.y * (16*NT);
    int mh = lane & 15;             // A row / B col within tile
    int kb = (lane >> 4) << 1;      // K sub-offset: lanes 0-15 -> {k,k+1}, 16-31 -> {k+2,k+3}
    v8f accv[NT] = {};
    const float* ar = A1 + (size_t)(row0 + mh) * K1;
    for (int kk = 0; kk < K1; kk += 4) {
        __builtin_prefetch(ar + kk + 128, 0, 1);   // stream-ahead on A row
        v2f av;
        av.x = ar[kk + kb];
        av.y = ar[kk + kb + 1];
        const float* wr0 = W1 + (size_t)(kk + kb) * ldW + col0 + mh;
        const float* wr1 = wr0 + ldW;
        #pragma unroll
        for (int t = 0; t < NT; ++t) {
            v2f bv;
            bv.x = wr0[t*16];
            bv.y = wr1[t*16];
            accv[t] = __builtin_amdgcn_wmma_f32_16x16x4_f32(false, av, false, bv,
                                                            (short)0, accv[t], false, false);
        }
    }
    if (A2) {
        const float* ar2 = A2 + (size_t)(row0 + mh) * K2;
        for (int kk = 0; kk < K2; kk += 4) {
            v2f av;
            av.x = ar2[kk + kb];
            av.y = ar2[kk + kb + 1];
            const float* wr0 = W2 + (size_t)(kk + kb) * ldW + col0 + mh;
            const float* wr1 = wr0 + ldW;
            #pragma unroll
            for (int t = 0; t < NT; ++t) {
                v2f bv;
                bv.x = wr0[t*16];
                bv.y = wr1[t*16];
                accv[t] = __builtin_amdgcn_wmma_f32_16x16x4_f32(false, av, false, bv,
                                                                (short)0, accv[t], false, false);
            }
        }
    }
    int half = lane >> 4;
    #pragma unroll
    for (int t = 0; t < NT; ++t) {
        #pragma unroll
        for (int r = 0; r < 8; ++r) {
            int m = r + half*8;     // C/D layout: VGPR r -> rows r / r+8
            int col = col0 + t*16 + mh;
            if (col < ncolsValid) {
                float v = accv[t][r];
                int oidx = (row0 + m) * ldOut + colOff + col;
                if (resid) v += resid[oidx];
                out[oidx] = v;
            }
        }
    }
}

// ---------------------------------------------------------------------------
// 6) magnitude activation: y = x * relu(|x|^2 + 1e-6 - SHIFT) / (|x|^2 + 1e-6)
// ---------------------------------------------------------------------------
__global__ void k_magact(const float* __restrict__ in, float* __restrict__ out, int C)
{
    int idx = blockIdx.x * blockDim.x + threadIdx.x;
    if (idx >= MROWS * C) return;
    const float* p = in  + (size_t)idx * 8;
    float*       q = out + (size_t)idx * 8;
    float x[8];
    float m2 = 1e-6f;
    #pragma unroll
    for (int m = 0; m < 8; ++m) { x[m] = p[m]; m2 += x[m]*x[m]; }
    float s = fmaxf(m2 - SHIFT_F, 0.0f) / m2;
    #pragma unroll
    for (int m = 0; m < 8; ++m) q[m] = x[m] * s;
}

// ---------------------------------------------------------------------------
// 7) finalize: p_corr, v_corr, m_matrix
// ---------------------------------------------------------------------------
__global__ void k_final(const float* __restrict__ L4out,  /* ld = 16, cols o*2+a */
                        const float* __restrict__ p1,
                        const float* __restrict__ p0,
                        float* __restrict__ dout)
{
    int bi = blockIdx.x * blockDim.x + threadIdx.x;
    if (bi >= MROWS) return;
    #pragma unroll
    for (int a2 = 0; a2 < 2; ++a2) {
        float pc = p1[bi*2 + a2] + CORR_F * L4out[bi*16 + a2];
        dout[bi*2 + a2]               = pc;                     // p_corr
        dout[MROWS*2 + bi*2 + a2]     = pc - p0[bi*2 + a2];     // v_corr (DT=1)
    }
    #pragma unroll
    for (int o = 1; o < 3; ++o)
        #pragma unroll
        for (int a2 = 0; a2 < 2; ++a2)
            dout[MROWS*4 + bi*4 + (o-1)*2 + a2] = CORR_F * L4out[bi*16 + o*2 + a2];
}

// ---------------------------------------------------------------------------
extern "C" void kernel_launch(void* const* d_in, const int* in_sizes, int n_in,
                              void* d_out, int out_size, void* d_ws, size_t ws_size,
                              hipStream_t stream)
{
    (void)in_sizes; (void)n_in; (void)out_size; (void)ws_size;
    const float* p0_enc = (const float*)d_in[0];
    const float* v0_enc = (const float*)d_in[1];
    const float* p0     = (const float*)d_in[2];
    const float* v0     = (const float*)d_in[3];
    const float* a      = (const float*)d_in[4];
    const float* mask   = (const float*)d_in[5];
    const float* lift_c0= (const float*)d_in[6];
    const float* Wc0    = (const float*)d_in[7];
    const float* lift_d0= (const float*)d_in[8];
    const float* Wd0    = (const float*)d_in[9];
    const float* Wc1    = (const float*)d_in[10];
    const float* Wd1    = (const float*)d_in[11];
    const float* Wc2    = (const float*)d_in[12];
    const float* Wd2    = (const float*)d_in[13];
    const float* Wc3    = (const float*)d_in[14];
    const float* Wd3    = (const float*)d_in[15];
    const float* Wc4    = (const float*)d_in[16];
    const float* proj_c4= (const float*)d_in[17];
    const float* Wd4    = (const float*)d_in[18];
    const float* proj_d4= (const float*)d_in[19];

    float* ws = (float*)d_ws;
    float* p1    = ws;                      // 2048
    float* denom = ws + 2048;               // 1024 (+ pad to 4096)
    float* Xc    = ws + 4096;               // 131072
    float* Xd    = Xc + 131072;
    float* Xact  = Xd + 131072;
    float* outA  = Xact + 131072;
    float* outB  = outA + 131072;
    float* Kmat  = outB + 131072;           // 786432 max
    float* Dmat  = Kmat + 786432;           // 16384
    unsigned int* gpk = (unsigned int*)(Dmat + 16384);   // 524288 uints
    float4* w4 = (float4*)(gpk + 524288);                // 524288 float4
    float*  F  = (float*)(w4 + 524288);                  // up to 6291456 floats

    // --- setup ---
    k_init_feats<<<64, 256, 0, stream>>>(p0_enc, v0_enc, p0, v0, a,
                                         lift_c0, lift_d0, p1, Xc, Xd);
    k_denom<<<4, 256, 0, stream>>>(p1, mask, denom);
    k_pairs<<<2048, 256, 0, stream>>>(p1, mask, denom, gpk, w4);

    // --- layer 0: conv(Xc) concat dense(Xd) -> outA (8 ch, ld 64) ---
    k_gather<<<MROWS, 128, 0, stream>>>(Xc, gpk, w4, F, 128);
    k_build_kmat<<<768, 256, 0, stream>>>(Wc0, Kmat, 16, 4);
    k_gemm<2><<<dim3(64, 1), 32, 0, stream>>>(F, Kmat, 6144, nullptr, nullptr, 0,
                                              nullptr, outA, 32, 32, 0, 64);
    k_build_dmat<<<16, 256, 0, stream>>>(Wd0, Dmat, 16, 4);
    k_gemm<2><<<dim3(64, 1), 32, 0, stream>>>(Xd, Dmat, 128, nullptr, nullptr, 0,
                                              nullptr, outA, 32, 32, 32, 64);

    // --- layer 1: 8 -> 8, residual ---
    k_magact<<<32, 256, 0, stream>>>(outA, Xact, 8);
    k_gather<<<MROWS, 128, 0, stream>>>(Xact, gpk, w4, F, 64);
    k_build_kmat<<<768, 256, 0, stream>>>(Wc1, Kmat, 8, 8);
    k_build_dmat<<<16, 256, 0, stream>>>(Wd1, Dmat, 8, 8);
    k_gemm<4><<<dim3(64, 1), 32, 0, stream>>>(F, Kmat, 3072, Xact, Dmat, 64,
                                              outA, outB, 64, 64, 0, 64);

    // --- layer 2: 8 -> 16 ---
    k_magact<<<32, 256, 0, stream>>>(outB, Xact, 8);
    k_gather<<<MROWS, 128, 0, stream>>>(Xact, gpk, w4, F, 64);
    k_build_kmat<<<1536, 256, 0, stream>>>(Wc2, Kmat, 8, 16);
    k_build_dmat<<<32, 256, 0, stream>>>(Wd2, Dmat, 8, 16);
    k_gemm<4><<<dim3(64, 2), 32, 0, stream>>>(F, Kmat, 3072, Xact, Dmat, 64,
                                              nullptr, outA, 128, 128, 0, 128);

    // --- layer 3: 16 -> 8 ---
    k_magact<<<64, 256, 0, stream>>>(outA, Xact, 16);
    k_gather<<<MROWS, 128, 0, stream>>>(Xact, gpk, w4, F, 128);
    k_build_kmat<<<1536, 256, 0, stream>>>(Wc3, Kmat, 16, 8);
    k_build_dmat<<<32, 256, 0, stream>>>(Wd3, Dmat, 16, 8);
    k_gemm<4><<<dim3(64, 1), 32, 0, stream>>>(F, Kmat, 6144, Xact, Dmat, 128,
                                              nullptr, outB, 64, 64, 0, 64);

    // --- layer 4: 8 -> 3, rho1 projection folded into weights (6 cols) ---
    k_magact<<<32, 256, 0, stream>>>(outB, Xact, 8);
    k_gather<<<MROWS, 128, 0, stream>>>(Xact, gpk, w4, F, 64);
    k_build_kmat4<<<192, 256, 0, stream>>>(Wc4, proj_c4, Kmat, 8, 3);
    k_build_dmat4<<<4, 256, 0, stream>>>(Wd4, proj_d4, Dmat, 8, 3);
    k_gemm<1><<<dim3(64, 1), 32, 0, stream>>>(F, Kmat, 3072, Xact, Dmat, 64,
                                              nullptr, outA, 16, 6, 0, 16);

    // --- outputs ---
    k_final<<<4, 256, 0, stream>>>(outA, p1, p0, (float*)d_out);
}